// train_GCN_2190433321519
// MI455X (gfx1250) — compile-verified
//
#include <hip/hip_runtime.h>
#include <hip/hip_bf16.h>

typedef float v2f __attribute__((ext_vector_type(2)));
typedef float v8f __attribute__((ext_vector_type(8)));

#define N_NODES  50000
#define N_EDGES  800000
#define N_GRAPHS 50
#define F_IN     128
#define F_MID    64
#define F_OUT    16

// ---------------------------------------------------------------------------
// In-degree (incl. one self loop per node), then dinv = rsqrt(deg) in place.
// ---------------------------------------------------------------------------
__global__ void deg_kernel(const int* __restrict__ col, float* __restrict__ deg) {
    int i = blockIdx.x * blockDim.x + threadIdx.x;
    if (i >= N_EDGES + N_NODES) return;
    int c = (i < N_EDGES) ? col[i] : (i - N_EDGES);
    atomicAdd(&deg[c], 1.0f);
}

__global__ void dinv_kernel(float* __restrict__ deg) {
    int i = blockIdx.x * blockDim.x + threadIdx.x;
    if (i >= N_NODES) return;
    float d = deg[i];
    deg[i] = (d > 0.0f) ? rsqrtf(d) : 0.0f;
}

// ---------------------------------------------------------------------------
// GEMM1: H1[N,64] = X[N,128] @ W1[128,64]  via V_WMMA_F32_16X16X4_F32.
// One 16-row tile per block; X tile staged in LDS (stride 132 -> 4-dword skew,
// conflict-free column access). 4 waves, each owns a 16-wide column slice.
// ---------------------------------------------------------------------------
__global__ __launch_bounds__(128) void gemm1_kernel(const float* __restrict__ X,
                                                    const float* __restrict__ W1,
                                                    float* __restrict__ H1) {
    __shared__ float As[16 * 132];
    const int rowBase = blockIdx.x * 16;
    const int t = threadIdx.x;

    // Coalesced staging of the 16x128 X tile.
#pragma unroll
    for (int i = 0; i < 16; ++i) {
        int idx = t + i * 128;          // 0..2047
        int m = idx >> 7;
        int k = idx & 127;
        As[m * 132 + k] = X[(rowBase + m) * F_IN + k];
    }
    __syncthreads();

    const int wave    = t >> 5;         // 0..3 -> output column tile
    const int lane    = t & 31;
    const int n       = lane & 15;      // A-row (M) and B-col (N) index
    const int kb      = (lane >> 4) << 1;  // lane-half selects K+0/K+2
    const int colBase = wave * 16;

    v8f acc = {0.f, 0.f, 0.f, 0.f, 0.f, 0.f, 0.f, 0.f};
#pragma unroll
    for (int k = 0; k < F_IN; k += 4) {
        v2f a, b;
        a.x = As[n * 132 + k + kb];
        a.y = As[n * 132 + k + kb + 1];
        b.x = W1[(k + kb)     * F_MID + colBase + n];
        b.y = W1[(k + kb + 1) * F_MID + colBase + n];
        acc = __builtin_amdgcn_wmma_f32_16x16x4_f32(
            false, a, false, b, (short)0, acc, false, false);
    }

    const int mrow = (lane >> 4) << 3;  // C layout: lanes 16-31 hold M=8..15
#pragma unroll
    for (int j = 0; j < 8; ++j)
        H1[(rowBase + mrow + j) * F_MID + colBase + n] = acc[j];
}

// ---------------------------------------------------------------------------
// GEMM2: H2[N,16] = relu(A1)[N,64] @ W2[64,16], ReLU fused into A-operand load.
// 4 waves per block, one 16-row tile each; A1 tile (4KB) is cache-resident.
// ---------------------------------------------------------------------------
__global__ __launch_bounds__(128) void gemm2_kernel(const float* __restrict__ A1,
                                                    const float* __restrict__ W2,
                                                    float* __restrict__ H2) {
    const int wave = threadIdx.x >> 5;
    const int lane = threadIdx.x & 31;
    const int rowTile = blockIdx.x * 4 + wave;
    if (rowTile * 16 >= N_NODES) return;   // uniform per wave: EXEC stays all-1
    const int rowBase = rowTile * 16;
    const int n  = lane & 15;
    const int kb = (lane >> 4) << 1;

    v8f acc = {0.f, 0.f, 0.f, 0.f, 0.f, 0.f, 0.f, 0.f};
#pragma unroll
    for (int k = 0; k < F_MID; k += 4) {
        v2f a, b;
        a.x = fmaxf(A1[(rowBase + n) * F_MID + k + kb],     0.0f);
        a.y = fmaxf(A1[(rowBase + n) * F_MID + k + kb + 1], 0.0f);
        b.x = W2[(k + kb)     * F_OUT + n];
        b.y = W2[(k + kb + 1) * F_OUT + n];
        acc = __builtin_amdgcn_wmma_f32_16x16x4_f32(
            false, a, false, b, (short)0, acc, false, false);
    }

    const int mrow = (lane >> 4) << 3;
#pragma unroll
    for (int j = 0; j < 8; ++j)
        H2[(rowBase + mrow + j) * F_OUT + n] = acc[j];
}

// ---------------------------------------------------------------------------
// Normalized scatter-aggregate: out[col] += H[row] * dinv[row]*dinv[col],
// plus the self-loop term. One thread per (edge, feature): consecutive
// threads share the edge -> coalesced gathers and coalesced f32 atomics.
// ---------------------------------------------------------------------------
template <int F>
__global__ void agg_kernel(const float* __restrict__ H,
                           const int* __restrict__ row,
                           const int* __restrict__ col,
                           const float* __restrict__ dinv,
                           float* __restrict__ out) {
    int idx = blockIdx.x * blockDim.x + threadIdx.x;
    if (idx >= (N_EDGES + N_NODES) * F) return;
    int e = idx / F;
    int f = idx % F;
    int r, c;
    if (e < N_EDGES) { r = row[e]; c = col[e]; }
    else             { r = c = e - N_EDGES; }          // self loop
    float nrm = dinv[r] * dinv[c];
    atomicAdd(&out[c * F + f], H[r * F + f] * nrm);
}

// ---------------------------------------------------------------------------
// Mean pool over sorted graph ids.
// ---------------------------------------------------------------------------
__global__ void pool_kernel(const float* __restrict__ A2,
                            const int* __restrict__ batch,
                            float* __restrict__ pooled,
                            float* __restrict__ counts) {
    int idx = blockIdx.x * blockDim.x + threadIdx.x;
    if (idx >= N_NODES * F_OUT) return;
    int node = idx >> 4;
    int f    = idx & 15;
    int g    = batch[node];
    atomicAdd(&pooled[g * F_OUT + f], A2[idx]);
    if (f == 0) atomicAdd(&counts[g], 1.0f);
}

__global__ void finalize_kernel(const float* __restrict__ pooled,
                                const float* __restrict__ counts,
                                float* __restrict__ out) {
    int idx = blockIdx.x * blockDim.x + threadIdx.x;
    if (idx >= N_GRAPHS * F_OUT) return;
    int g = idx >> 4;
    out[idx] = pooled[idx] / fmaxf(counts[g], 1.0f);
}

// ---------------------------------------------------------------------------
// Workspace layout (floats):
//   [deg/dinv 50000][A1 3.2M][A2 0.8M][pooled 800][counts 64] <- zeroed
//   [H1 3.2M][H2 0.8M]                                        <- fully written
// Total ~32.2 MB.
// ---------------------------------------------------------------------------
extern "C" void kernel_launch(void* const* d_in, const int* in_sizes, int n_in,
                              void* d_out, int out_size, void* d_ws, size_t ws_size,
                              hipStream_t stream) {
    const float* x     = (const float*)d_in[0];
    const float* W1    = (const float*)d_in[1];
    const float* W2    = (const float*)d_in[2];
    const int*   ei    = (const int*)d_in[3];
    const int*   batch = (const int*)d_in[4];
    const int* row = ei;
    const int* col = ei + N_EDGES;

    float* ws     = (float*)d_ws;
    float* deg    = ws;                 // 50000 (becomes dinv)
    float* A1     = ws + 50048;         // 3,200,000
    float* A2     = ws + 3250048;       // 800,000
    float* pooled = ws + 4050048;       // 800
    float* counts = ws + 4050848;       // 64 (padded)
    float* H1     = ws + 4050944;       // 3,200,000
    float* H2     = ws + 7250944;       // 800,000

    // Zero all accumulation buffers (deg..counts are contiguous).
    hipMemsetAsync(ws, 0, (size_t)4050912 * sizeof(float), stream);

    deg_kernel<<<(N_EDGES + N_NODES + 255) / 256, 256, 0, stream>>>(col, deg);
    dinv_kernel<<<(N_NODES + 255) / 256, 256, 0, stream>>>(deg);

    gemm1_kernel<<<N_NODES / 16, 128, 0, stream>>>(x, W1, H1);

    agg_kernel<F_MID><<<((N_EDGES + N_NODES) * F_MID + 255) / 256, 256, 0, stream>>>(
        H1, row, col, deg, A1);

    gemm2_kernel<<<(N_NODES / 16 + 3) / 4, 128, 0, stream>>>(A1, W2, H2);

    agg_kernel<F_OUT><<<((N_EDGES + N_NODES) * F_OUT + 255) / 256, 256, 0, stream>>>(
        H2, row, col, deg, A2);

    pool_kernel<<<(N_NODES * F_OUT + 255) / 256, 256, 0, stream>>>(A2, batch, pooled, counts);
    finalize_kernel<<<(N_GRAPHS * F_OUT + 255) / 256, 256, 0, stream>>>(pooled, counts,
                                                                        (float*)d_out);
}